// DynamicHypergraphConv_19215683682841
// MI455X (gfx1250) — compile-verified
//
#include <hip/hip_runtime.h>

// ---------------------------------------------------------------------------
// DynamicHypergraphConv for MI455X (gfx1250), wave32 + V_WMMA_F32_16X16X4_F32
// ---------------------------------------------------------------------------

typedef float v2f __attribute__((ext_vector_type(2)));
typedef float v8f __attribute__((ext_vector_type(8)));

#define NROWS 8192
#define DIM   256
#define TOPK  10

// ---------------------------------------------------------------------------
// fp32 WMMA GEMM with compile-time K/N: C[M x NN] = A[M x KD] * B[KD x NN]
// (+ bias). One 16x16 tile per wave, 8 waves / block. Constexpr strides ->
// all inner-loop addresses are base + immediate offset.
// ---------------------------------------------------------------------------
template <int KD, int NN>
__global__ __launch_bounds__(256) void gemm_f32_wmma(
    const float* __restrict__ A, const float* __restrict__ B,
    const float* __restrict__ bias, float* __restrict__ C)
{
    const int lane = threadIdx.x & 31;
    const int wave = threadIdx.x >> 5;
    const int tile = blockIdx.x * 8 + wave;
    constexpr int ntn = NN / 16;
    const int ti = tile / ntn;            // 16-row tile index
    const int tj = tile % ntn;            // 16-col tile index

    const int khalf = (lane >> 4) * 2;    // lanes 16-31 hold K+2,K+3
    const float* Abase = A + (ti * 16 + (lane & 15)) * KD + khalf;
    const float* Bbase = B + khalf * NN + tj * 16 + (lane & 15);

    v8f c = {};
    #pragma unroll 8
    for (int k0 = 0; k0 < KD; k0 += 4) {
        v2f a = *(const v2f*)(Abase + k0);  // contiguous K pair, imm offset
        v2f b;
        b.x = Bbase[k0 * NN];               // constexpr stride -> imm offset
        b.y = Bbase[(k0 + 1) * NN];
        c = __builtin_amdgcn_wmma_f32_16x16x4_f32(
                false, a, false, b, (short)0, c, false, false);
    }

    const float bv = bias ? bias[tj * 16 + (lane & 15)] : 0.0f;
    float* Cbase = C + (ti * 16 + ((lane >> 4) ? 8 : 0)) * NN
                     + tj * 16 + (lane & 15);
    #pragma unroll
    for (int r = 0; r < 8; ++r)
        Cbase[r * NN] = c[r] + bv;
}

// ---------------------------------------------------------------------------
// Row squared norms of f
// ---------------------------------------------------------------------------
__global__ __launch_bounds__(256) void rownorm_kernel(
    const float* __restrict__ f, float* __restrict__ sq)
{
    __shared__ float red[8];
    const int i = blockIdx.x;
    float v = f[i * DIM + threadIdx.x];
    float s = v * v;
    #pragma unroll
    for (int o = 16; o > 0; o >>= 1) s += __shfl_down(s, o, 32);
    if ((threadIdx.x & 31) == 0) red[threadIdx.x >> 5] = s;
    __syncthreads();
    if (threadIdx.x == 0) {
        float t = 0.0f;
        #pragma unroll
        for (int w = 0; w < 8; ++w) t += red[w];
        sq[i] = t;
    }
}

// ---------------------------------------------------------------------------
// Fused Gram matrix + streaming per-row top-K (kNN).
// Block = 256 threads = 8 waves, owns 64 rows. Per jb iteration the block
// covers a 64x32 score tile (waves arranged 4 row-subtiles x 2 col-subtiles).
// score(i,j) = sq[j] - 2*<f_i, f_j>   (rank-equivalent to cdist; sq[i], sqrt
// dropped as rank-invariant per row). Distance matrix never hits HBM.
//
// The block's 64-row A panel of f (64 KB) is staged ONCE into LDS and shared
// by all 8 waves for the whole 8192-column sweep: inner loop per WMMA is one
// ds_load_b64 (A frag) + one global_load_b64 (B frag), both base+imm offset.
// DScnt/LOADcnt are separate counters so A and B loads overlap.
// ---------------------------------------------------------------------------
__global__ __launch_bounds__(256) void gram_topk_kernel(
    const float* __restrict__ f, const float* __restrict__ sq,
    int* __restrict__ idx, int* __restrict__ bdeg)
{
    __shared__ float sA[64 * DIM];            // 64 KB A panel
    __shared__ float sdist[64 * 32];          // 8 KB score tile
    __shared__ float mD[64 * 4 * TOPK];       // merge buffers
    __shared__ int   mI[64 * 4 * TOPK];

    const int lane = threadIdx.x & 31;
    const int wave = threadIdx.x >> 5;
    const int rb   = blockIdx.x * 64;
    const int mt   = wave & 3;            // row subtile 0..3
    const int nt   = wave >> 2;           // col subtile 0..1

    // ---- stage the 64 x 256 A panel into LDS (coalesced float4 copy)
    {
        const float4* src = (const float4*)(f + rb * DIM);
        float4*       dst = (float4*)sA;
        #pragma unroll
        for (int c = 0; c < (64 * DIM / 4) / 256; ++c)
            dst[c * 256 + threadIdx.x] = src[c * 256 + threadIdx.x];
    }
    __syncthreads();

    const int khalf = (lane >> 4) * 2;
    const float* Alds = sA + (mt * 16 + (lane & 15)) * DIM + khalf;

    // private sorted (ascending) top-K for this thread's candidate stream
    float bd[TOPK];
    int   bi[TOPK];
    #pragma unroll
    for (int k = 0; k < TOPK; ++k) { bd[k] = 3.0e38f; bi[k] = -1; }
    const int trow = threadIdx.x & 63;        // scan-phase row in block
    const int tcol = (threadIdx.x >> 6) * 8;  // scan-phase col offset

    for (int jb = 0; jb < NROWS; jb += 32) {
        const int bcol = jb + nt * 16 + (lane & 15);
        const float* Brow = f + bcol * DIM + khalf;
        // prefetch this lane's B row for the NEXT column block into caches
        __builtin_prefetch(Brow + 32 * DIM, 0, 1);

        v8f c = {};
        #pragma unroll 8
        for (int k0 = 0; k0 < DIM / 4; ++k0) {
            v2f a = *(const v2f*)(Alds + k0 * 4);   // ds_load_b64, imm offset
            v2f b = *(const v2f*)(Brow + k0 * 4);   // global_load_b64, imm off
            c = __builtin_amdgcn_wmma_f32_16x16x4_f32(
                    false, a, false, b, (short)0, c, false, false);
        }

        const float sqj = sq[bcol];
        #pragma unroll
        for (int r = 0; r < 8; ++r) {
            const int lr = mt * 16 + r + ((lane >> 4) ? 8 : 0);
            const int lc = nt * 16 + (lane & 15);
            sdist[lr * 32 + lc] = sqj - 2.0f * c[r];
        }
        __syncthreads();

        // each thread scans 8 fresh candidates for its row
        #pragma unroll
        for (int q = 0; q < 8; ++q) {
            const float d = sdist[trow * 32 + tcol + q];
            if (d < bd[TOPK - 1]) {
                const int j = jb + tcol + q;
                int p = TOPK - 1;
                while (p > 0 && bd[p - 1] > d) {
                    bd[p] = bd[p - 1]; bi[p] = bi[p - 1]; --p;
                }
                bd[p] = d; bi[p] = j;
            }
        }
        __syncthreads();
    }

    // merge the 4 partial sorted lists per row
    const int part = threadIdx.x >> 6;
    #pragma unroll
    for (int k = 0; k < TOPK; ++k) {
        mD[(trow * 4 + part) * TOPK + k] = bd[k];
        mI[(trow * 4 + part) * TOPK + k] = bi[k];
    }
    __syncthreads();

    if (threadIdx.x < 64) {
        const int row = rb + threadIdx.x;
        const float* pd = &mD[threadIdx.x * 4 * TOPK];
        const int*   pi = &mI[threadIdx.x * 4 * TOPK];
        int ptr[4] = {0, 0, 0, 0};
        for (int k = 0; k < TOPK; ++k) {          // 4-way sorted merge
            float best = 3.1e38f; int bp = 0;
            #pragma unroll
            for (int p = 0; p < 4; ++p) {
                if (ptr[p] < TOPK) {
                    const float v = pd[p * TOPK + ptr[p]];
                    if (v < best) { best = v; bp = p; }
                }
            }
            const int j = pi[bp * TOPK + ptr[bp]];
            ptr[bp]++;
            idx[row * TOPK + k] = j;
            atomicAdd(&bdeg[j], 1);               // hyperedge degree
        }
    }
}

// ---------------------------------------------------------------------------
// u = H^T @ x  (scatter, HW fp32 atomics)
// ---------------------------------------------------------------------------
__global__ __launch_bounds__(256) void scatter_kernel(
    const float* __restrict__ x, const int* __restrict__ idx,
    float* __restrict__ u)
{
    const int i = blockIdx.x;
    const int d = threadIdx.x;
    const float v = x[i * DIM + d];
    #pragma unroll
    for (int k = 0; k < TOPK; ++k) {
        const int j = idx[i * TOPK + k];
        unsafeAtomicAdd(&u[j * DIM + d], v);
    }
}

// ---------------------------------------------------------------------------
// h = x - 0.5 * c2 * H @ (b_is * u), with c2 = d_is^2 = 1/(10+1e-10)
// ---------------------------------------------------------------------------
__global__ __launch_bounds__(256) void gather_kernel(
    const float* __restrict__ x, const int* __restrict__ idx,
    const int* __restrict__ bdeg, const float* __restrict__ u,
    float* __restrict__ h)
{
    const int i = blockIdx.x;
    const int d = threadIdx.x;
    const float c2 = 1.0f / (10.0f + 1e-10f);
    float m = 0.0f;
    #pragma unroll
    for (int k = 0; k < TOPK; ++k) {
        const int j = idx[i * TOPK + k];
        const float bis = rsqrtf((float)bdeg[j] + 1e-10f);
        m += bis * u[j * DIM + d];
    }
    h[i * DIM + d] = x[i * DIM + d] - 0.5f * c2 * m;
}

// ---------------------------------------------------------------------------
extern "C" void kernel_launch(void* const* d_in, const int* in_sizes, int n_in,
                              void* d_out, int out_size, void* d_ws, size_t ws_size,
                              hipStream_t stream)
{
    const float* x    = (const float*)d_in[0];   // (8192, 256)
    const float* w    = (const float*)d_in[1];   // (256, 256)
    const float* wdyn = (const float*)d_in[2];   // (256, 256)
    const float* bias = (const float*)d_in[3];   // (256,)
    float* out = (float*)d_out;

    char* ws = (char*)d_ws;
    float* f    = (float*)ws;  ws += (size_t)NROWS * DIM * sizeof(float);  // reused as h
    float* u    = (float*)ws;  ws += (size_t)NROWS * DIM * sizeof(float);
    float* sq   = (float*)ws;  ws += (size_t)NROWS * sizeof(float);
    int*   idx  = (int*)ws;    ws += (size_t)NROWS * TOPK * sizeof(int);
    int*   bdeg = (int*)ws;    ws += (size_t)NROWS * sizeof(int);

    hipMemsetAsync(u, 0, (size_t)NROWS * DIM * sizeof(float), stream);
    hipMemsetAsync(bdeg, 0, (size_t)NROWS * sizeof(int), stream);

    const int gemm_blocks = (NROWS / 16) * (DIM / 16) / 8;   // 1024

    // 1) f = x @ weight_dynamic
    gemm_f32_wmma<DIM, DIM><<<gemm_blocks, 256, 0, stream>>>(x, wdyn, nullptr, f);
    // 2) sq[i] = ||f_i||^2
    rownorm_kernel<<<NROWS, 256, 0, stream>>>(f, sq);
    // 3) fused Gram + kNN top-10 + hyperedge degrees
    gram_topk_kernel<<<NROWS / 64, 256, 0, stream>>>(f, sq, idx, bdeg);
    // 4) u = H^T @ x
    scatter_kernel<<<NROWS, 256, 0, stream>>>(x, idx, u);
    // 5) h = x - 0.5*c2*H @ (b_is * u)   (writes into f buffer)
    gather_kernel<<<NROWS, 256, 0, stream>>>(x, idx, bdeg, u, f);
    // 6) out = h @ weight + bias
    gemm_f32_wmma<DIM, DIM><<<gemm_blocks, 256, 0, stream>>>(f, w, bias, out);
}